// WeightedMeanSAGELayer_81836306858014
// MI455X (gfx1250) — compile-verified
//
#include <hip/hip_runtime.h>
#include <hip/hip_bf16.h>

typedef float v2f __attribute__((ext_vector_type(2)));
typedef float v8f __attribute__((ext_vector_type(8)));

#define IN_FEAT 128
#define OUT_FEAT 128
#define SCAN_THREADS 256

// ---------------------------------------------------------------------------
// Phase 1a: per-node in-degree count (int atomics; 1 per edge)
// ---------------------------------------------------------------------------
__global__ __launch_bounds__(256) void sage_count_kernel(
    const int* __restrict__ dst, int* __restrict__ cnt, int n_edges)
{
    int e = blockIdx.x * 256 + threadIdx.x;
    if (e < n_edges) atomicAdd(&cnt[dst[e]], 1);
}

// ---------------------------------------------------------------------------
// Phase 1b: single-block exclusive prefix sum over n_nodes counts.
// Each thread serially sums a chunk, block-scan of chunk totals in LDS,
// then writes exclusive offsets for its chunk.  off has n_nodes+1 entries.
// ---------------------------------------------------------------------------
__global__ __launch_bounds__(SCAN_THREADS) void sage_scan_kernel(
    const int* __restrict__ cnt, int* __restrict__ off, int n_nodes)
{
    __shared__ int sums[SCAN_THREADS];
    const int chunk = (n_nodes + SCAN_THREADS - 1) / SCAN_THREADS;
    const int t     = threadIdx.x;
    const int begin = t * chunk;
    const int end   = min(begin + chunk, n_nodes);

    int local = 0;
    for (int i = begin; i < end; ++i) local += cnt[i];

    int x = local;
    sums[t] = x;
    __syncthreads();
    for (int s = 1; s < SCAN_THREADS; s <<= 1) {
        int y = (t >= s) ? sums[t - s] : 0;
        __syncthreads();
        x += y;
        sums[t] = x;
        __syncthreads();
    }
    int run = x - local;  // exclusive prefix of this chunk
    for (int i = begin; i < end; ++i) {
        off[i] = run;
        run += cnt[i];
    }
    if (t == SCAN_THREADS - 1) off[n_nodes] = run;
}

// ---------------------------------------------------------------------------
// Phase 1c: CSR fill (1 int atomic per edge)
// ---------------------------------------------------------------------------
__global__ __launch_bounds__(256) void sage_fill_kernel(
    const int* __restrict__ dst, const int* __restrict__ off,
    int* __restrict__ cursor, int* __restrict__ csr, int n_edges)
{
    int e = blockIdx.x * 256 + threadIdx.x;
    if (e >= n_edges) return;
    int d = dst[e];
    int p = atomicAdd(&cursor[d], 1);
    csr[off[d] + p] = e;
}

// ---------------------------------------------------------------------------
// Phase 2: gather-aggregate.  One wave per node; each lane owns a float4
// slice of the 128-float feature row.  Pure reads (L2-resident), register
// accumulation, no feature atomics.  hN = sum / max(deg,1).
// ---------------------------------------------------------------------------
__global__ __launch_bounds__(256) void sage_aggregate_kernel(
    const float* __restrict__ h, const float* __restrict__ w,
    const int* __restrict__ src, const int* __restrict__ off,
    const int* __restrict__ csr, float* __restrict__ hN, int n_nodes)
{
    const int wave = threadIdx.x >> 5;
    const int lane = threadIdx.x & 31;
    const int node = blockIdx.x * 8 + wave;
    if (node >= n_nodes) return;

    const int j0 = off[node];
    const int j1 = off[node + 1];

    float4 acc = {0.f, 0.f, 0.f, 0.f};
    for (int j = j0; j < j1; ++j) {
        const int e = csr[j];          // uniform within wave
        const int s = src[e];          // uniform within wave
        const float we = w[e];
        const float4* __restrict__ hrow = (const float4*)(h + (size_t)s * IN_FEAT);
        float4 v = hrow[lane];
        acc.x += v.x * we;
        acc.y += v.y * we;
        acc.z += v.z * we;
        acc.w += v.w * we;
    }
    const float dg = fmaxf((float)(j1 - j0), 1.0f);
    float4 r = {acc.x / dg, acc.y / dg, acc.z / dg, acc.w / dg};
    float4* __restrict__ out = (float4*)(hN + (size_t)node * IN_FEAT);
    out[lane] = r;
}

// ---------------------------------------------------------------------------
// Phase 3: GEMM  out = [h | h_N] @ W + b   via V_WMMA_F32_16X16X4_F32.
// One wave computes one 16x16 output tile; K = 256 in steps of 4.
//
// Fragment layouts (CDNA5 ISA 7.12.2, 32-bit, wave32):
//   A (16x4, MxK):  lane L, vgpr v -> A[row = L&15][k = 2*(L>>4) + v]
//   B (4x16, KxN):  lane L, vgpr v -> B[k = 2*(L>>4) + v][col = L&15]
//   C/D (16x16):    lane L, vgpr r -> D[row = r + 8*(L>>4)][col = L&15]
// ---------------------------------------------------------------------------
__global__ __launch_bounds__(256) void sage_gemm_kernel(
    const float* __restrict__ h,     // [N,128]  (cols 0..127 of h_total)
    const float* __restrict__ hN,    // [N,128]  (cols 128..255 of h_total)
    const float* __restrict__ W,     // [256,128]
    const float* __restrict__ bias,  // [128]
    float* __restrict__ out)         // [N,128]
{
    const int wave = threadIdx.x >> 5;
    const int lane = threadIdx.x & 31;
    const int tile = blockIdx.x * 8 + wave;   // 8 waves/block, 1 tile/wave

    const int mtile = tile >> 3;
    const int ntile = tile & 7;

    const int m0  = mtile * 16;
    const int n0  = ntile * 16;
    const int lo  = lane & 15;                // A row index = B/D column index
    const int kh2 = (lane >> 4) * 2;          // K sub-offset {0,2}

    const float* __restrict__ hrow  = h  + (size_t)(m0 + lo) * IN_FEAT;
    const float* __restrict__ hNrow = hN + (size_t)(m0 + lo) * IN_FEAT;
    const int ncol = n0 + lo;

    v8f acc = {};

    // K = 0..127 : A from h
    for (int k = 0; k < 128; k += 4) {
        const int ka = k + kh2;
        v2f a, bm;
        a[0]  = hrow[ka];
        a[1]  = hrow[ka + 1];
        bm[0] = W[(size_t)ka * OUT_FEAT + ncol];
        bm[1] = W[(size_t)(ka + 1) * OUT_FEAT + ncol];
        acc = __builtin_amdgcn_wmma_f32_16x16x4_f32(
            false, a, false, bm, (short)0, acc, false, false);
    }
    // K = 128..255 : A from h_N, B from second half of W
    for (int k = 0; k < 128; k += 4) {
        const int ka = k + kh2;
        v2f a, bm;
        a[0]  = hNrow[ka];
        a[1]  = hNrow[ka + 1];
        bm[0] = W[(size_t)(128 + ka) * OUT_FEAT + ncol];
        bm[1] = W[(size_t)(128 + ka + 1) * OUT_FEAT + ncol];
        acc = __builtin_amdgcn_wmma_f32_16x16x4_f32(
            false, a, false, bm, (short)0, acc, false, false);
    }

    const float bv = bias[ncol];
    const int rbase = m0 + (lane >> 4) * 8;
    for (int r = 0; r < 8; ++r) {
        out[(size_t)(rbase + r) * OUT_FEAT + ncol] = acc[r] + bv;
    }
}

// ---------------------------------------------------------------------------
// Launch: inputs in setup_inputs() order: h, w, W, b, src, dst
// ---------------------------------------------------------------------------
extern "C" void kernel_launch(void* const* d_in, const int* in_sizes, int n_in,
                              void* d_out, int out_size, void* d_ws, size_t ws_size,
                              hipStream_t stream) {
    const float* h    = (const float*)d_in[0];
    const float* w    = (const float*)d_in[1];
    const float* W    = (const float*)d_in[2];
    const float* bias = (const float*)d_in[3];
    const int*   src  = (const int*)d_in[4];
    const int*   dst  = (const int*)d_in[5];
    float* out = (float*)d_out;

    const int n_nodes = in_sizes[0] / IN_FEAT;   // 10000
    const int n_edges = in_sizes[4];             // 640000

    // Workspace layout:
    //   [hN : n_nodes*128 f32][off : n_nodes+1 i32][cnt : n_nodes i32]
    //   [cursor : n_nodes i32][csr : n_edges i32]
    float* hN   = (float*)d_ws;
    int*   offp = (int*)(hN + (size_t)n_nodes * IN_FEAT);
    int*   cnt  = offp + (n_nodes + 1);
    int*   cur  = cnt + n_nodes;
    int*   csr  = cur + n_nodes;

    // zero the two counter arrays (contiguous: cnt then cur)
    hipMemsetAsync(cnt, 0, (size_t)2 * n_nodes * sizeof(int), stream);

    const int eblocks = (n_edges + 255) / 256;
    sage_count_kernel<<<eblocks, 256, 0, stream>>>(dst, cnt, n_edges);
    sage_scan_kernel<<<1, SCAN_THREADS, 0, stream>>>(cnt, offp, n_nodes);
    sage_fill_kernel<<<eblocks, 256, 0, stream>>>(dst, offp, cur, csr, n_edges);

    const int nblocks = (n_nodes + 7) / 8;       // one wave per node
    sage_aggregate_kernel<<<nblocks, 256, 0, stream>>>(
        h, w, src, offp, csr, hN, n_nodes);

    // (n_nodes/16) * (128/16) tiles, 8 tiles (waves) per block
    const int tiles = (n_nodes / 16) * (OUT_FEAT / 16);
    sage_gemm_kernel<<<tiles / 8, 256, 0, stream>>>(h, hN, W, bias, out);
}